// VDELoss_27092653703433
// MI455X (gfx1250) — compile-verified
//
#include <hip/hip_runtime.h>
#include <math.h>

// Problem constants (fixed by the reference)
#define N_ROWS 65536
#define D_OUT  128
#define D_LAT  64
#define D_Z    64

// Grid configuration
#define NB_R 256
#define NT_R 256
#define NB_K 128
#define NT_K 256
#define NB_Z 64            // must be multiple of 4 (WMMA K-tiles of 4 rows)
#define NT_Z 1024

// Workspace layout (in floats)
#define WS_RECON 0                       // [NB_R]
#define WS_KL    (WS_RECON + NB_R)       // [NB_K]
#define WS_CROSS (WS_KL + NB_K)          // [NB_Z]
#define WS_P     (WS_CROSS + NB_Z)       // P[q][NB_Z][64], q: 0=S_t 1=S_tau 2=Q_t 3=Q_tau

typedef __attribute__((ext_vector_type(2))) float v2f;
typedef __attribute__((ext_vector_type(8))) float v8f;

// ---------------------------------------------------------------------------
// Kernel 1: reconstruction MSE partials.  Streams 64 MB with b128 loads.
// ---------------------------------------------------------------------------
__global__ void recon_partial(const float* __restrict__ target,
                              const float* __restrict__ output,
                              float* __restrict__ part) {
    __shared__ float sh[NT_R];
    const int t = threadIdx.x;
    const long n4 = (long)N_ROWS * D_OUT / 4;
    const long stride = (long)gridDim.x * NT_R;
    const float4* t4 = (const float4*)target;
    const float4* o4 = (const float4*)output;
    float acc = 0.f;
    for (long i = (long)blockIdx.x * NT_R + t; i < n4; i += stride) {
        float4 a = o4[i], b = t4[i];
        float dx = a.x - b.x, dy = a.y - b.y, dz = a.z - b.z, dw = a.w - b.w;
        acc += dx * dx + dy * dy + dz * dz + dw * dw;
    }
    sh[t] = acc;
    __syncthreads();
    for (int s = NT_R / 2; s > 0; s >>= 1) {
        if (t < s) sh[t] += sh[t + s];
        __syncthreads();
    }
    if (t == 0) part[blockIdx.x] = sh[0];
}

// ---------------------------------------------------------------------------
// Kernel 2: KL partials.  term = lv - exp(lv) - mean^2 + 1, summed flat.
// ---------------------------------------------------------------------------
__global__ void kl_partial(const float* __restrict__ mean,
                           const float* __restrict__ logv,
                           float* __restrict__ part) {
    __shared__ float sh[NT_K];
    const int t = threadIdx.x;
    const long n4 = (long)N_ROWS * D_LAT / 4;
    const long stride = (long)gridDim.x * NT_K;
    const float4* m4 = (const float4*)mean;
    const float4* l4 = (const float4*)logv;
    float acc = 0.f;
    for (long i = (long)blockIdx.x * NT_K + t; i < n4; i += stride) {
        float4 m = m4[i], l = l4[i];
        acc += (l.x - expf(l.x) - m.x * m.x + 1.f);
        acc += (l.y - expf(l.y) - m.y * m.y + 1.f);
        acc += (l.z - expf(l.z) - m.z * m.z + 1.f);
        acc += (l.w - expf(l.w) - m.w * m.w + 1.f);
    }
    sh[t] = acc;
    __syncthreads();
    for (int s = NT_K / 2; s > 0; s >>= 1) {
        if (t < s) sh[t] += sh[t + s];
        __syncthreads();
    }
    if (t == 0) part[blockIdx.x] = sh[0];
}

// ---------------------------------------------------------------------------
// Kernel 3: z statistics.  Per-block: per-column partial S_t, S_tau, Q_t,
// Q_tau (64 cols) + scalar cross partial.  1024 threads: thread t owns column
// group 4*(t%16) (one float4) of row rowBase + iter*64 + t/16 -> fully
// coalesced b128 streams over both z arrays.
// ---------------------------------------------------------------------------
__global__ void zstats_partial(const float* __restrict__ zt,
                               const float* __restrict__ ztau,
                               float* __restrict__ ws) {
    __shared__ float4 sh4[NT_Z];
    __shared__ float  shc[NT_Z];
    const int t = threadIdx.x;
    const int g = t & 15;        // float4 column group (cols 4g..4g+3)
    const int rl = t >> 4;       // 0..63 local row
    const int rowsPerBlock = N_ROWS / NB_Z;          // 1024
    const int rowBase = blockIdx.x * rowsPerBlock;

    float4 st = {0, 0, 0, 0}, stau = {0, 0, 0, 0};
    float4 qt = {0, 0, 0, 0}, qtau = {0, 0, 0, 0};
    float cross = 0.f;

    #pragma unroll 4
    for (int it = 0; it < rowsPerBlock / 64; ++it) {  // 16 iters of 64 rows
        const long r = rowBase + it * 64 + rl;
        const float4 a = *(const float4*)(zt   + r * D_Z + 4 * g);
        const float4 b = *(const float4*)(ztau + r * D_Z + 4 * g);
        st.x += a.x; st.y += a.y; st.z += a.z; st.w += a.w;
        stau.x += b.x; stau.y += b.y; stau.z += b.z; stau.w += b.w;
        qt.x += a.x * a.x; qt.y += a.y * a.y; qt.z += a.z * a.z; qt.w += a.w * a.w;
        qtau.x += b.x * b.x; qtau.y += b.y * b.y; qtau.z += b.z * b.z; qtau.w += b.w * b.w;
        cross += a.x * b.x + a.y * b.y + a.z * b.z + a.w * b.w;
    }

    // per-column reduction: column j gathered from the 64 threads with
    // t%16 == j/4, component j%4.  Deterministic fixed-order loop.
    float4 accs[4] = {st, stau, qt, qtau};
    float* P = ws + WS_P;
    for (int q = 0; q < 4; ++q) {
        sh4[t] = accs[q];
        __syncthreads();
        if (t < 64) {
            const int grp = t >> 2, comp = t & 3;
            float s = 0.f;
            for (int i = 0; i < 64; ++i) {
                const float* f = (const float*)&sh4[i * 16 + grp];
                s += f[comp];
            }
            P[((long)q * NB_Z + blockIdx.x) * 64 + t] = s;
        }
        __syncthreads();
    }

    // scalar cross partial
    shc[t] = cross;
    __syncthreads();
    for (int s = NT_Z / 2; s > 0; s >>= 1) {
        if (t < s) shc[t] += shc[t + s];
        __syncthreads();
    }
    if (t == 0) ws[WS_CROSS + blockIdx.x] = shc[0];
}

// ---------------------------------------------------------------------------
// Kernel 4: finalize.  16 waves each reduce one (quantity, 16-column group)
// slice of the 64x64 partials matrix with V_WMMA_F32_16X16X4_F32 using an
// all-ones A(16x4): every row of D = column sums of the accumulated B tiles.
// A==ones makes the result invariant to the exact K-slot layout of B; the
// N = lane%16 mapping matches the documented C/D layout.
// ---------------------------------------------------------------------------
__global__ void finalize(const float* __restrict__ ws, float* __restrict__ out) {
    __shared__ float colsum[4][D_Z];
    const int t = threadIdx.x;
    const int wave = t >> 5, lane = t & 31;
    const int q = wave >> 2, g = wave & 3;          // 16 waves = 4 quantities x 4 col groups
    const float* Pq = ws + WS_P + (long)q * NB_Z * 64;
    const int col = g * 16 + (lane & 15);
    const int rb = (lane < 16) ? 0 : 2;             // lanes 0-15: K rows {b,b+1}; 16-31: {b+2,b+3}

    v2f a; a.x = 1.f; a.y = 1.f;                    // ones A-matrix (16x4 f32, 2 VGPRs)
    v8f c = {};                                     // f32 16x16 accumulator (8 VGPRs)
    for (int b = 0; b < NB_Z; b += 4) {
        v2f bt;
        bt.x = Pq[(long)(b + rb) * 64 + col];
        bt.y = Pq[(long)(b + rb + 1) * 64 + col];
        c = __builtin_amdgcn_wmma_f32_16x16x4_f32(false, a, false, bt,
                                                  (short)0, c, false, false);
    }
    if (lane < 16) colsum[q][g * 16 + lane] = c[0]; // every D row identical; take M=0
    __syncthreads();

    if (t == 0) {
        double rsum = 0.0, ksum = 0.0, csum = 0.0;
        for (int i = 0; i < NB_R; ++i) rsum += ws[WS_RECON + i];
        for (int i = 0; i < NB_K; ++i) ksum += ws[WS_KL + i];
        for (int i = 0; i < NB_Z; ++i) csum += ws[WS_CROSS + i];
        const double Nf = (double)N_ROWS;

        const double recon = rsum / (Nf * (double)D_OUT);
        const double kl = -0.5 * ksum / Nf;

        double denom = 0.0, mudot = 0.0;
        for (int j = 0; j < D_Z; ++j) {
            const double St = colsum[0][j], Stau = colsum[1][j];
            const double Qt = colsum[2][j], Qtau = colsum[3][j];
            const double vt   = (Qt - St * St / Nf) / (Nf - 1.0);
            const double vtau = (Qtau - Stau * Stau / Nf) / (Nf - 1.0);
            denom += sqrt(vt) * sqrt(vtau);        // dot(std_t, std_tau)
            mudot += St * Stau;
        }
        const double diagSum = csum - mudot / Nf;  // sum_i diag
        const double autoLoss = -(diagSum / Nf) / denom;

        out[0] = (float)(recon + kl + autoLoss);
    }
}

// ---------------------------------------------------------------------------
extern "C" void kernel_launch(void* const* d_in, const int* in_sizes, int n_in,
                              void* d_out, int out_size, void* d_ws, size_t ws_size,
                              hipStream_t stream) {
    const float* target = (const float*)d_in[0];
    const float* output = (const float*)d_in[1];
    const float* mean   = (const float*)d_in[2];
    const float* logv   = (const float*)d_in[3];
    const float* zt     = (const float*)d_in[4];
    const float* ztau   = (const float*)d_in[5];
    float* ws  = (float*)d_ws;
    float* out = (float*)d_out;

    recon_partial<<<NB_R, NT_R, 0, stream>>>(target, output, ws + WS_RECON);
    kl_partial<<<NB_K, NT_K, 0, stream>>>(mean, logv, ws + WS_KL);
    zstats_partial<<<NB_Z, NT_Z, 0, stream>>>(zt, ztau, ws);
    finalize<<<1, 512, 0, stream>>>(ws, out);
}